// RVQ_69312182223727
// MI455X (gfx1250) — compile-verified
//
#include <hip/hip_runtime.h>
#include <math.h>

typedef __attribute__((ext_vector_type(2))) float v2f;
typedef __attribute__((ext_vector_type(8))) float v8f;

#define NSTAGE 8
#define NCODE  1024
#define DIM    256
#define NROWS  131072
#define WPB    4          // waves per block (wave32)

// ---------------------------------------------------------------------------
// Prep: cnorm[s*1024+k] = ||codebook[s][k]||^2 ; initialize used_out from input
// ---------------------------------------------------------------------------
__global__ __launch_bounds__(256) void rvq_prep(const float* __restrict__ cb,
                                                const int*   __restrict__ cb_used,
                                                float*       __restrict__ cnorm,
                                                int*         __restrict__ used_out) {
    int i = blockIdx.x * blockDim.x + threadIdx.x;   // 0 .. 8191
    const float* c = cb + (size_t)i * DIM;
    float s = 0.f;
#pragma unroll 8
    for (int d = 0; d < DIM; d += 4) {
        float4 v = *(const float4*)(c + d);
        s = fmaf(v.x, v.x, s); s = fmaf(v.y, v.y, s);
        s = fmaf(v.z, v.z, s); s = fmaf(v.w, v.w, s);
    }
    cnorm[i]    = s;
    used_out[i] = cb_used[i];
}

// ---------------------------------------------------------------------------
// Main: one wave per 16-row tile; residual lives in WMMA-A fragment registers
// across all 8 stages. Score GEMM via V_WMMA_F32_16X16X4_F32.
// ---------------------------------------------------------------------------
__global__ __launch_bounds__(WPB * 32) void rvq_main(
        const float* __restrict__ input,
        const float* __restrict__ cb,
        const float* __restrict__ noise,
        const int*   __restrict__ cb_used,
        const int*   __restrict__ train_mode,
        const float* __restrict__ cnorm,
        float*       __restrict__ out,
        int*         __restrict__ used_out) {

    __shared__ int s_idx[WPB][16];

    const int lane = threadIdx.x & 31;
    const int w    = threadIdx.x >> 5;
    const int m    = lane & 15;        // row (A) or col (B/C) within tile
    const int hh   = lane >> 4;        // half-wave: selects K pair {2hh, 2hh+1}
    const long long wid  = (long long)blockIdx.x * WPB + w;
    const long long row0 = wid * 16;

    // Residual in A-fragment layout: lane holds row m, dims 4*kb + 2*hh + {0,1}
    const float* abase = input + (size_t)(row0 + m) * DIM + 2 * hh;
    float resid[128];
#pragma unroll
    for (int kb = 0; kb < 64; ++kb) {
        v2f v = *(const v2f*)(abase + 4 * kb);
        resid[2 * kb]     = v.x;
        resid[2 * kb + 1] = v.y;
    }

    for (int stage = 0; stage < NSTAGE; ++stage) {
        const float* cbs = cb + (size_t)stage * NCODE * DIM;
        const float* cns = cnorm + stage * NCODE;

        float bestD[8];
        int   bestI[8];
#pragma unroll
        for (int j = 0; j < 8; ++j) { bestD[j] = 3.0e38f; bestI[j] = 0x7fffffff; }

        for (int nt = 0; nt < NCODE / 16; ++nt) {
            // B fragment: lane holds code (nt*16+m), dims 4*kb + 2*hh + {0,1}
            const float* bbase = cbs + (size_t)(nt * 16 + m) * DIM + 2 * hh;
            v8f c = {0.f, 0.f, 0.f, 0.f, 0.f, 0.f, 0.f, 0.f};
#pragma unroll
            for (int kb = 0; kb < 64; ++kb) {
                v2f a; a.x = resid[2 * kb]; a.y = resid[2 * kb + 1];
                v2f b = *(const v2f*)(bbase + 4 * kb);
                c = __builtin_amdgcn_wmma_f32_16x16x4_f32(
                        false, a, false, b, (short)0, c, false, false);
            }
            // dist = ||c||^2 - 2*dot  (row-constant ||r||^2 dropped)
            float cn  = cns[nt * 16 + m];
            int   idx = nt * 16 + m;
#pragma unroll
            for (int j = 0; j < 8; ++j) {
                float d = fmaf(-2.f, c[j], cn);
                if (d < bestD[j]) { bestD[j] = d; bestI[j] = idx; }  // first-min kept
            }
        }

        // Cross-lane argmin within each 16-lane half (row j+8*hh), tie -> low idx
#pragma unroll
        for (int j = 0; j < 8; ++j) {
            float d = bestD[j]; int idx = bestI[j];
#pragma unroll
            for (int off = 1; off < 16; off <<= 1) {
                float d2 = __shfl_xor(d,   off, 32);
                int   i2 = __shfl_xor(idx, off, 32);
                if (d2 < d || (d2 == d && i2 < idx)) { d = d2; idx = i2; }
            }
            bestD[j] = d; bestI[j] = idx;
        }

        // Broadcast per-row chosen code via per-wave LDS slot
        if (lane == 0) {
#pragma unroll
            for (int j = 0; j < 8; ++j) s_idx[w][j] = bestI[j];
        }
        if (lane == 16) {
#pragma unroll
            for (int j = 0; j < 8; ++j) s_idx[w][8 + j] = bestI[j];
        }
        __syncthreads();
        int myidx = s_idx[w][m];
        __syncthreads();

        // residual -= codebook[myidx]  (gathered b64 loads, same fragment layout)
        const float* cvec = cbs + (size_t)myidx * DIM + 2 * hh;
#pragma unroll
        for (int kb = 0; kb < 64; ++kb) {
            v2f v = *(const v2f*)(cvec + 4 * kb);
            resid[2 * kb]     -= v.x;
            resid[2 * kb + 1] -= v.y;
        }
        // presence update: same-value races across rows/blocks are benign
        if (lane < 16)
            used_out[stage * NCODE + myidx] = cb_used[stage * NCODE + myidx] + 1;
    }

    // err^2 per row: lanes m and m+16 jointly cover all 256 dims of row m
    float e2 = 0.f;
#pragma unroll
    for (int i = 0; i < 128; ++i) e2 = fmaf(resid[i], resid[i], e2);
    e2 += __shfl_xor(e2, 16, 32);

    const float* nbase = noise + (size_t)(row0 + m) * DIM + 2 * hh;
    float n2 = 0.f;
#pragma unroll
    for (int kb = 0; kb < 64; ++kb) {
        v2f v = *(const v2f*)(nbase + 4 * kb);
        n2 = fmaf(v.x, v.x, n2); n2 = fmaf(v.y, v.y, n2);
    }
    n2 += __shfl_xor(n2, 16, 32);

    float* obase = out + (size_t)(row0 + m) * DIM + 2 * hh;
    if (*train_mode) {
        float s = sqrtf(e2) / sqrtf(n2) + 1e-12f;
#pragma unroll
        for (int kb = 0; kb < 64; ++kb) {
            v2f a  = *(const v2f*)(abase + 4 * kb);
            v2f nv = *(const v2f*)(nbase + 4 * kb);
            v2f o; o.x = fmaf(s, nv.x, a.x); o.y = fmaf(s, nv.y, a.y);
            *(v2f*)(obase + 4 * kb) = o;
        }
    } else {
        // total_q = input - final_residual
#pragma unroll
        for (int kb = 0; kb < 64; ++kb) {
            v2f a = *(const v2f*)(abase + 4 * kb);
            v2f o; o.x = a.x - resid[2 * kb]; o.y = a.y - resid[2 * kb + 1];
            *(v2f*)(obase + 4 * kb) = o;
        }
    }
}

// ---------------------------------------------------------------------------
extern "C" void kernel_launch(void* const* d_in, const int* in_sizes, int n_in,
                              void* d_out, int out_size, void* d_ws, size_t ws_size,
                              hipStream_t stream) {
    const float* input   = (const float*)d_in[0];
    const float* cb      = (const float*)d_in[1];
    const float* noise   = (const float*)d_in[2];
    const int*   cb_used = (const int*)d_in[3];
    const int*   train   = (const int*)d_in[4];

    float* out      = (float*)d_out;
    int*   used_out = (int*)((float*)d_out + (size_t)NROWS * DIM);
    float* cnorm    = (float*)d_ws;   // 8192 floats

    rvq_prep<<<(NSTAGE * NCODE) / 256, 256, 0, stream>>>(cb, cb_used, cnorm, used_out);

    int blocks = (NROWS / 16) / WPB;  // 2048 blocks of 4 waves
    rvq_main<<<blocks, WPB * 32, 0, stream>>>(input, cb, noise, cb_used, train,
                                              cnorm, out, used_out);
}